// HashTexel_13271448945251
// MI455X (gfx1250) — compile-verified
//
#include <hip/hip_runtime.h>
#include <stdint.h>

typedef float v2f __attribute__((ext_vector_type(2)));
typedef float v4f __attribute__((ext_vector_type(4)));
typedef int   v4i __attribute__((ext_vector_type(4)));

#define N_LEVELS 16
#define LOG2_T   19
#define TSIZE    (1 << LOG2_T)          // 524288 entries per level
#define NPTS     (64 * 64 * 128)        // 524288 points
#define BLOCK    256

// LDS staging layout for dense levels 0..2 (res = 16, 32, 64)
//  level0: 256  entries * 2 f32 =  512 floats  @ offset 0
//  level1: 1024 entries * 2 f32 = 2048 floats  @ offset 512
//  level2: 4096 entries * 2 f32 = 8192 floats  @ offset 2560
#define SMEM_FLOATS 10752               // 43008 bytes

#if __has_builtin(__builtin_amdgcn_global_load_async_to_lds_b128)
#define USE_ASYNC_LDS 1
#else
#define USE_ASYNC_LDS 0
#endif

// Address-space-qualified vector pointers for the async-to-LDS builtin:
// param0 = int4 addrspace(1)* (global), param1 = int4 addrspace(3)* (LDS)
typedef __attribute__((address_space(1))) v4i g_v4i;
typedef __attribute__((address_space(3))) v4i l_v4i;

__device__ __forceinline__ void stage_level(const float* __restrict__ gsrc,
                                            float* ldst, int nf4, int tid) {
#if USE_ASYNC_LDS
  g_v4i* g = (g_v4i*)(const void*)gsrc;
  l_v4i* l = (l_v4i*)(void*)ldst;
  for (int i = tid; i < nf4; i += BLOCK) {
    __builtin_amdgcn_global_load_async_to_lds_b128(g + i, l + i, 0, 0);
  }
#else
  const v4f* g = (const v4f*)gsrc;
  v4f* l = (v4f*)ldst;
  for (int i = tid; i < nf4; i += BLOCK) l[i] = g[i];
#endif
}

__global__ __launch_bounds__(BLOCK)
void hashgrid_encode_kernel(const float* __restrict__ x,
                            const float* __restrict__ table,
                            float* __restrict__ out) {
  __shared__ float smem[SMEM_FLOATS];
  const int tid = threadIdx.x;

  // ---- Stage dense levels 0..2 into LDS (async DMA path on CDNA5) ----
  stage_level(table + (size_t)0 * TSIZE * 2, smem + 0,    128,  tid);
  stage_level(table + (size_t)1 * TSIZE * 2, smem + 512,  512,  tid);
  stage_level(table + (size_t)2 * TSIZE * 2, smem + 2560, 2048, tid);
#if USE_ASYNC_LDS
#if __has_builtin(__builtin_amdgcn_s_wait_asynccnt)
  __builtin_amdgcn_s_wait_asynccnt(0);
#else
  asm volatile("s_wait_asynccnt 0" ::: "memory");
#endif
#endif
  __syncthreads();

  const int p = blockIdx.x * BLOCK + tid;
  if (p >= NPTS) return;

  // coords in [-1,1] -> [0,1]
  v2f xy = ((const v2f*)x)[p];
  const float x0 = xy.x * 0.5f + 0.5f;
  const float y0 = xy.y * 0.5f + 0.5f;

  float o[2 * N_LEVELS];
  const int lds_base[3] = {0, 512, 2560};

#pragma unroll
  for (int l = 0; l < N_LEVELS; ++l) {
    const int res = 16 << l;                         // 2^l * base_res
    const float scale = (float)res - 1.0f;           // grid scale
    // dense (non-hashed) iff res^2 <= T  -> levels 0..5
    const bool dense = ((long long)res * (long long)res <= (long long)TSIZE);

    const float px = x0 * scale + 0.5f;
    const float py = y0 * scale + 0.5f;
    const float fx = floorf(px), fy = floorf(py);
    const float tx = px - fx, ty = py - fy;
    const int ix = (int)fx, iy = (int)fy;
    // smoothstep weights
    const float wx = tx * tx * (3.0f - 2.0f * tx);
    const float wy = ty * ty * (3.0f - 2.0f * ty);

    const float* __restrict__ tbl = table + (size_t)l * TSIZE * 2;

    float f0 = 0.0f, f1 = 0.0f;
#pragma unroll
    for (int c = 0; c < 4; ++c) {                    // tcnn corner order: dim0 fastest
      const int dx = c & 1, dy = c >> 1;
      const unsigned cx = (unsigned)(ix + dx);
      const unsigned cy = (unsigned)(iy + dy);
      unsigned idx;
      if (dense) {
        const unsigned mask = (unsigned)res * (unsigned)res - 1u;  // res^2 is pow2
        idx = (cx + cy * (unsigned)res) & mask;
      } else {
        idx = (cx * 1u) ^ (cy * 2654435761u);
        idx &= (unsigned)(TSIZE - 1);
      }
      const float w = (dx ? wx : 1.0f - wx) * (dy ? wy : 1.0f - wy);
      v2f f;
      if (l < 3) {
        f = *(const v2f*)(smem + lds_base[l] + 2 * idx);   // ds_load_b64
      } else {
        f = *(const v2f*)(tbl + 2 * (size_t)idx);          // global_load_b64 (L2-resident)
      }
      f0 = fmaf(w, f.x, f0);
      f1 = fmaf(w, f.y, f1);
    }
    o[2 * l + 0] = f0;
    o[2 * l + 1] = f1;
  }

  // 32 contiguous f32 per point; non-temporal b128 stores (don't pollute L2,
  // keep the 64MB hash table resident).
  float* op = out + (size_t)p * (2 * N_LEVELS);
#pragma unroll
  for (int i = 0; i < 8; ++i) {
    v4f v = {o[4 * i + 0], o[4 * i + 1], o[4 * i + 2], o[4 * i + 3]};
    __builtin_nontemporal_store(v, (v4f*)(op + 4 * i));
  }
}

extern "C" void kernel_launch(void* const* d_in, const int* in_sizes, int n_in,
                              void* d_out, int out_size, void* d_ws, size_t ws_size,
                              hipStream_t stream) {
  const float* x     = (const float*)d_in[0];   // [64,64,128,2] f32
  const float* table = (const float*)d_in[1];   // [16, 2^19, 2] f32
  float* out = (float*)d_out;                   // [64,64,128,32] f32
  (void)in_sizes; (void)n_in; (void)out_size; (void)d_ws; (void)ws_size;

  hashgrid_encode_kernel<<<NPTS / BLOCK, BLOCK, 0, stream>>>(x, table, out);
}